// GCN_54537494724630
// MI455X (gfx1250) — compile-verified
//
#include <hip/hip_runtime.h>
#include <hip/hip_bf16.h>

// ---------------------------------------------------------------------------
// 2-layer GCN for MI455X (gfx1250, wave32).
//   out = dinv[i] * (sum_{e:dst=i} hs[src_e] + hs[i]) + b,  hs = (X@W)*dinv[row]
// GEMM uses V_WMMA_F32_16X16X4_F32 (fp32-exact); scatter uses native fp32
// global atomics; degree via fp32 atomics (exact for counts < 2^24).
// ---------------------------------------------------------------------------

typedef __attribute__((ext_vector_type(2))) float v2f;
typedef __attribute__((ext_vector_type(8))) float v8f;

#define IN_C  128
#define HID_C 128
#define OUT_C 64
#define LDA   132   // LDS row stride (132 % 64 = 4 -> conflict-free, float4-aligned)

// ---- simple fill -----------------------------------------------------------
__global__ void gcn_fill(float* __restrict__ p, float v, int n) {
    int t = blockIdx.x * blockDim.x + threadIdx.x;
    if (t < n) p[t] = v;
}

// ---- degree count: deg[dst[e]] += 1 ---------------------------------------
__global__ void gcn_deg_count(const int* __restrict__ dst, float* __restrict__ deg, int E) {
    int t = blockIdx.x * blockDim.x + threadIdx.x;
    if (t < E)
        __hip_atomic_fetch_add(&deg[dst[t]], 1.0f, __ATOMIC_RELAXED, __HIP_MEMORY_SCOPE_AGENT);
}

// ---- dinv = rsqrt(deg) in place (deg >= 1 always: self loop) --------------
__global__ void gcn_rsqrt(float* __restrict__ d, int n) {
    int t = blockIdx.x * blockDim.x + threadIdx.x;
    if (t < n) d[t] = rsqrtf(d[t]);
}

// ---- WMMA GEMM + row scale: HS[m, :] = (X[m, :] @ W) * dinv[m] ------------
// One 16-row tile per block; wave w owns column tile w (16 cols).
// K = 128 fixed. blockDim.x = 32 * (NCOLS/16).
template <int NCOLS>
__global__ void gcn_gemm_scale(const float* __restrict__ X,
                               const float* __restrict__ W,
                               const float* __restrict__ dinv,
                               float* __restrict__ HS,
                               int N) {
    __shared__ float As[16 * LDA];

    const int row0 = blockIdx.x * 16;
    const int tid  = threadIdx.x;

    // Cooperative load of 16 x 128 A tile (zero-pad rows >= N).
    for (int i = tid; i < 16 * IN_C; i += blockDim.x) {
        int r = i >> 7;          // / 128
        int k = i & (IN_C - 1);  // % 128
        int m = row0 + r;
        As[r * LDA + k] = (m < N) ? X[(size_t)m * IN_C + k] : 0.0f;
    }
    __syncthreads();

    const int wave = tid >> 5;
    const int lane = tid & 31;
    const int ln   = lane & 15;   // N column within tile / M row for A frag
    const int lh   = lane >> 4;   // half select (K offset 2*lh)
    const int col0 = wave * 16;

    v8f acc = {};
#pragma unroll
    for (int k0 = 0; k0 < IN_C; k0 += 4) {
        // A fragment 16x4: lane%16 = M row; VGPR0 holds K = k0+2*lh, VGPR1 next K.
        v2f a;
        a.x = As[ln * LDA + k0 + 2 * lh];
        a.y = As[ln * LDA + k0 + 2 * lh + 1];
        // B fragment 4x16: lane%16 = N col; VGPR0 holds K = k0+2*lh, VGPR1 next K.
        v2f b;
        b.x = W[(size_t)(k0 + 2 * lh)     * NCOLS + col0 + ln];
        b.y = W[(size_t)(k0 + 2 * lh + 1) * NCOLS + col0 + ln];
        acc = __builtin_amdgcn_wmma_f32_16x16x4_f32(
                  /*neg_a=*/false, a, /*neg_b=*/false, b,
                  /*c_mod=*/(short)0, acc, /*reuse_a=*/false, /*reuse_b=*/false);
    }

    // D layout: VGPR r -> M = r + 8*(lane>=16), N = lane%16.
#pragma unroll
    for (int r = 0; r < 8; ++r) {
        int m = row0 + r + 8 * lh;
        if (m < N)
            HS[(size_t)m * NCOLS + col0 + ln] = acc[r] * dinv[m];
    }
}

// ---- edge scatter: agg[dst, :] += hs[src, :]  (float4 per thread) ---------
template <int C>
__global__ void gcn_scatter(const int* __restrict__ src, const int* __restrict__ dst,
                            const float* __restrict__ hs, float* __restrict__ agg, int E) {
    constexpr int C4 = C / 4;
    int t = blockIdx.x * blockDim.x + threadIdx.x;
    if (t >= E * C4) return;
    int e = t / C4;
    int g = t - e * C4;
    int s = src[e];
    int d = dst[e];
    float4 v = reinterpret_cast<const float4*>(hs)[(size_t)s * C4 + g];
    float* o = agg + (size_t)d * C + 4 * g;
    __hip_atomic_fetch_add(o + 0, v.x, __ATOMIC_RELAXED, __HIP_MEMORY_SCOPE_AGENT);
    __hip_atomic_fetch_add(o + 1, v.y, __ATOMIC_RELAXED, __HIP_MEMORY_SCOPE_AGENT);
    __hip_atomic_fetch_add(o + 2, v.z, __ATOMIC_RELAXED, __HIP_MEMORY_SCOPE_AGENT);
    __hip_atomic_fetch_add(o + 3, v.w, __ATOMIC_RELAXED, __HIP_MEMORY_SCOPE_AGENT);
}

// ---- finalize (in place on agg): agg = [relu](dinv*(agg + hs) + b) --------
template <int C, bool RELU>
__global__ void gcn_finalize(float* __restrict__ agg, const float* __restrict__ hs,
                             const float* __restrict__ dinv, const float* __restrict__ b,
                             int n4) {
    constexpr int C4 = C / 4;
    int t = blockIdx.x * blockDim.x + threadIdx.x;
    if (t >= n4) return;
    int node = t / C4;
    int g    = t - node * C4;
    float4 a  = reinterpret_cast<float4*>(agg)[t];
    float4 h  = reinterpret_cast<const float4*>(hs)[t];
    float4 bb = reinterpret_cast<const float4*>(b)[g];
    float  di = dinv[node];
    float4 o;
    o.x = di * (a.x + h.x) + bb.x;
    o.y = di * (a.y + h.y) + bb.y;
    o.z = di * (a.z + h.z) + bb.z;
    o.w = di * (a.w + h.w) + bb.w;
    if (RELU) {
        o.x = fmaxf(o.x, 0.0f); o.y = fmaxf(o.y, 0.0f);
        o.z = fmaxf(o.z, 0.0f); o.w = fmaxf(o.w, 0.0f);
    }
    reinterpret_cast<float4*>(agg)[t] = o;
}

static inline int cdiv(int a, int b) { return (a + b - 1) / b; }

extern "C" void kernel_launch(void* const* d_in, const int* in_sizes, int n_in,
                              void* d_out, int out_size, void* d_ws, size_t ws_size,
                              hipStream_t stream) {
    const float* x  = (const float*)d_in[0];
    const int*   ei = (const int*)d_in[1];
    const float* W1 = (const float*)d_in[2];
    const float* b1 = (const float*)d_in[3];
    const float* W2 = (const float*)d_in[4];
    const float* b2 = (const float*)d_in[5];
    float* out = (float*)d_out;

    const int N = in_sizes[0] / IN_C;   // 50000
    const int E = in_sizes[1] / 2;      // 800000
    const int* src = ei;
    const int* dst = ei + E;

    // Workspace layout (floats): dinv | hs1 (N*128) | agg1 (N*128) | hs2 (N*64)
    float* ws    = (float*)d_ws;
    const size_t nPad = ((size_t)N + 127) & ~(size_t)127;
    float* dinv = ws;
    float* hs1  = dinv + nPad;
    float* agg1 = hs1 + (size_t)N * HID_C;
    float* hs2  = agg1 + (size_t)N * HID_C;

    const int THREADS = 256;

    // 1) degree = 1 (self loop) + edge counts; dinv = rsqrt(deg)
    gcn_fill<<<cdiv(N, THREADS), THREADS, 0, stream>>>(dinv, 1.0f, N);
    gcn_deg_count<<<cdiv(E, THREADS), THREADS, 0, stream>>>(dst, dinv, E);
    gcn_rsqrt<<<cdiv(N, THREADS), THREADS, 0, stream>>>(dinv, N);

    const int rowTiles = cdiv(N, 16);   // 3125

    // 2) layer 1: hs1 = (x @ W1) * dinv
    gcn_gemm_scale<HID_C><<<rowTiles, 32 * (HID_C / 16), 0, stream>>>(x, W1, dinv, hs1, N);
    // 3) agg1 = 0; scatter edges
    gcn_fill<<<cdiv(N * HID_C, THREADS), THREADS, 0, stream>>>(agg1, 0.0f, N * HID_C);
    gcn_scatter<HID_C><<<cdiv(E * (HID_C / 4), THREADS), THREADS, 0, stream>>>(src, dst, hs1, agg1, E);
    // 4) agg1 = relu(dinv*(agg1 + hs1) + b1)
    gcn_finalize<HID_C, true><<<cdiv(N * HID_C / 4, THREADS), THREADS, 0, stream>>>(agg1, hs1, dinv, b1, N * HID_C / 4);

    // 5) layer 2: hs2 = (agg1 @ W2) * dinv
    gcn_gemm_scale<OUT_C><<<rowTiles, 32 * (OUT_C / 16), 0, stream>>>(agg1, W2, dinv, hs2, N);
    // 6) d_out as accumulator: zero, scatter, finalize (no relu)
    gcn_fill<<<cdiv(N * OUT_C, THREADS), THREADS, 0, stream>>>(out, 0.0f, N * OUT_C);
    gcn_scatter<OUT_C><<<cdiv(E * (OUT_C / 4), THREADS), THREADS, 0, stream>>>(src, dst, hs2, out, E);
    gcn_finalize<OUT_C, false><<<cdiv(N * OUT_C / 4, THREADS), THREADS, 0, stream>>>(out, hs2, dinv, b2, N * OUT_C / 4);
}